// EngineGNN_39256001086004
// MI455X (gfx1250) — compile-verified
//
#include <hip/hip_runtime.h>
#include <math.h>

typedef __attribute__((ext_vector_type(2))) float v2f;
typedef __attribute__((ext_vector_type(8))) float v8f;

#define TPB 256

// ---------------------------------------------------------------- init ws
__global__ void init_ws(int* __restrict__ first, float* __restrict__ S,
                        float* __restrict__ cnt, int nT, int nTor) {
  int i = blockIdx.x * blockDim.x + threadIdx.x;
  if (i < nT) first[i] = 0x7FFFFFFF;        // segment_min fill (> any edge id)
  if (i < nTor * 3) S[i] = 0.0f;
  if (i < nTor) cnt[i] = 0.0f;
}

// ------------------------------------------------------- x_t = x_terrain
__global__ void copy_x(const float4* __restrict__ x, float4* __restrict__ xt,
                       int nT) {
  int i = blockIdx.x * blockDim.x + threadIdx.x;
  if (i < nT) xt[i] = x[i];
}

// ------------------------------------------- SequenceConv via f32 WMMA
// One wave handles 16 edges. A = 16 gathered source rows (16x4 f32),
// B = W_seq^T broadcast into columns 0..3 of a 4x16 (rest zero),
// D[m][j] = msg[eBase+m][j]; scatter-added with global f32 atomics.
// Tail handling is branchless: out-of-range slots are clamped to edge E-1
// and contribute exactly 0.0f (their A rows are zeroed before the WMMA).
__global__ void seq_conv(const float* __restrict__ x, const int* __restrict__ src,
                         const int* __restrict__ dst, const float* __restrict__ W,
                         float* __restrict__ xt, int E) {
  const int lane = threadIdx.x & 31;
  const int wave = blockIdx.x * (blockDim.x >> 5) + (threadIdx.x >> 5);
  const int eBase = wave * 16;
  if (eBase >= E) return;          // wave-uniform: EXEC all-ones at the WMMA

  const int m = lane & 15;         // A: edge slot; B/D: column index
  const int half = lane >> 4;      // 0 -> K={0,1}, 1 -> K={2,3}
  const int rem = E - eBase;       // wave-uniform

  // A-matrix: 16x4, lanes 0-15 K=0,1 ; lanes 16-31 K=2,3 (ISA layout)
  int e = eBase + m;
  int ec = e < E ? e : (E - 1);                   // clamped, always valid
  const float* row = x + (unsigned)src[ec] * 4u + half * 2;
  float ax = row[0];
  float ay = row[1];
  bool av = m < rem;
  v2f a;
  a.x = av ? ax : 0.0f;                           // v_cndmask, no EXEC games
  a.y = av ? ay : 0.0f;
  // B-matrix: B[k][n] = W_seq[n][k] for n<4, zero otherwise
  int wn = m & 3;
  float bx = W[wn * 4 + half * 2 + 0];
  float by = W[wn * 4 + half * 2 + 1];
  bool bv = m < 4;
  v2f b;
  b.x = bv ? bx : 0.0f;
  b.y = bv ? by : 0.0f;

  v8f c = {};
  c = __builtin_amdgcn_wmma_f32_16x16x4_f32(false, a, false, b, (short)0, c,
                                            false, false);
  // D layout: c[r] at lane L = D[r + 8*(L>>4)][L&15]
  if (m < 4) {
    const int tbase = eBase + 8 * half;   // edges covered by this lane's column
#pragma unroll
    for (int r = 0; r < 8; ++r) {
      int ee = tbase + r;
      int eec = ee < E ? ee : (E - 1);    // clamped; c[r] == 0 when ee >= E
      atomicAdd(&xt[(unsigned)dst[eec] * 4u + m], c[r]);
    }
  }
}

// ------------------------------------------- first assignment edge / node
__global__ void assign_first(const int* __restrict__ asrc, int* __restrict__ first,
                             int E) {
  int e = blockIdx.x * blockDim.x + threadIdx.x;
  if (e < E) atomicMin(&first[asrc[e]], e);
}

// ------------------------------------------- per-node operator transform
__global__ void op_transform(float* __restrict__ xt, const int* __restrict__ first,
                             const int* __restrict__ adst,
                             const float* __restrict__ pol,
                             const float* __restrict__ opW,
                             const float* __restrict__ opb, int nT, int E) {
  int n = blockIdx.x * blockDim.x + threadIdx.x;
  if (n >= nT) return;
  int f = first[n];
  if (f >= E) return;                         // no assignment: keep x_t
  float4 xr = ((const float4*)xt)[n];
  int op = adst[f];
  op = op < 0 ? 0 : (op > 3 ? 3 : op);
  float p = pol[f];
  const float* Wp = opW + op * 20;            // op_W[op] is 4x5 row-major
  const float* bp = opb + op * 4;
  float4 o;
  o.x = Wp[0]  * xr.x + Wp[1]  * xr.y + Wp[2]  * xr.z + Wp[3]  * xr.w + Wp[4]  * p + bp[0];
  o.y = Wp[5]  * xr.x + Wp[6]  * xr.y + Wp[7]  * xr.z + Wp[8]  * xr.w + Wp[9]  * p + bp[1];
  o.z = Wp[10] * xr.x + Wp[11] * xr.y + Wp[12] * xr.z + Wp[13] * xr.w + Wp[14] * p + bp[2];
  o.w = Wp[15] * xr.x + Wp[16] * xr.y + Wp[17] * xr.z + Wp[18] * xr.w + Wp[19] * p + bp[3];
  ((float4*)xt)[n] = o;
}

// --------------------------------------------- TransportConv edge phase
__global__ void transport_edges(const float* __restrict__ xtor,
                                const int* __restrict__ tr,
                                const float* __restrict__ gW,
                                const float* __restrict__ gb,
                                float* __restrict__ S, float* __restrict__ cnt,
                                int E) {
  int e = blockIdx.x * blockDim.x + threadIdx.x;
  if (e >= E) return;
  unsigned s = (unsigned)tr[e];          // tr_ei[0][e]
  unsigned d = (unsigned)tr[E + e];      // tr_ei[1][e]
  const float* Xi = xtor + d * 3u;
  const float* Xj = xtor + s * 3u;
  float xi0 = Xi[0], xi1 = Xi[1], xi2 = Xi[2];
  float xj0 = Xj[0], xj1 = Xj[1], xj2 = Xj[2];
  float z = xi0 * gW[0] + xi1 * gW[1] + xi2 * gW[2] +
            xj0 * gW[3] + xj1 * gW[4] + xj2 * gW[5] + gb[0];
  float g = 1.0f / (1.0f + expf(-z));
  atomicAdd(&S[d * 3u + 0u], g * xj0);
  atomicAdd(&S[d * 3u + 1u], g * xj1);
  atomicAdd(&S[d * 3u + 2u], g * xj2);
  atomicAdd(&cnt[d], 1.0f);
}

// ------------------------------------------- TransportConv node finalize
__global__ void transport_finalize(const float* __restrict__ xtor,
                                   const float* __restrict__ S,
                                   const float* __restrict__ cnt,
                                   float* __restrict__ out, int nTor) {
  int n = blockIdx.x * blockDim.x + threadIdx.x;
  if (n >= nTor) return;
  float c = fmaxf(cnt[n], 1.0f);
  unsigned b = (unsigned)n * 3u;
  out[b + 0u] = xtor[b + 0u] + S[b + 0u] / c;
  out[b + 1u] = xtor[b + 1u] + S[b + 1u] / c;
  out[b + 2u] = xtor[b + 2u] + S[b + 2u] / c;
}

static inline size_t align16(size_t x) { return (x + 15) & ~(size_t)15; }

extern "C" void kernel_launch(void* const* d_in, const int* in_sizes, int n_in,
                              void* d_out, int out_size, void* d_ws, size_t ws_size,
                              hipStream_t stream) {
  const float* x_terrain  = (const float*)d_in[0];
  const float* polarity   = (const float*)d_in[1];
  const float* x_torus    = (const float*)d_in[2];
  const int*   seq_ei     = (const int*)d_in[3];
  const int*   assign_src = (const int*)d_in[4];
  const int*   assign_dst = (const int*)d_in[5];
  const int*   tr_ei      = (const int*)d_in[6];
  const float* W_seq      = (const float*)d_in[7];
  const float* op_W       = (const float*)d_in[8];
  const float* op_b       = (const float*)d_in[9];
  const float* gate_W     = (const float*)d_in[10];
  const float* gate_b     = (const float*)d_in[11];

  const int N_T   = in_sizes[0] / 4;
  const int E_SEQ = in_sizes[3] / 2;
  const int E_AS  = in_sizes[4];
  const int N_TOR = in_sizes[2] / 3;
  const int E_TR  = in_sizes[6] / 2;

  float* xt       = (float*)d_out;                       // x_t: [N_T, 4]
  float* xtor_out = (float*)d_out + (size_t)N_T * 4;     // x_tor: [N_TOR, 3]

  char* ws    = (char*)d_ws;
  int*  first = (int*)ws;
  float* S    = (float*)(ws + align16((size_t)N_T * sizeof(int)));
  float* cnt  = (float*)((char*)S + align16((size_t)N_TOR * 3 * sizeof(float)));

  const int* seq_src = seq_ei;
  const int* seq_dst = seq_ei + E_SEQ;

  // 1) init workspace
  {
    int total = N_T > 3 * N_TOR ? N_T : 3 * N_TOR;
    init_ws<<<(total + TPB - 1) / TPB, TPB, 0, stream>>>(first, S, cnt, N_T, N_TOR);
  }
  // 2) x_t = x_terrain
  copy_x<<<(N_T + TPB - 1) / TPB, TPB, 0, stream>>>(
      (const float4*)x_terrain, (float4*)xt, N_T);
  // 3) SequenceConv: WMMA edge messages + scatter-add residual
  {
    long waves = ((long)E_SEQ + 15) / 16;
    long threads = waves * 32;
    int blocks = (int)((threads + TPB - 1) / TPB);
    seq_conv<<<blocks, TPB, 0, stream>>>(x_terrain, seq_src, seq_dst, W_seq, xt, E_SEQ);
  }
  // 4) first assignment edge per node (segment_min)
  assign_first<<<(E_AS + TPB - 1) / TPB, TPB, 0, stream>>>(assign_src, first, E_AS);
  // 5) per-node operator transform (in-place on d_out)
  op_transform<<<(N_T + TPB - 1) / TPB, TPB, 0, stream>>>(
      xt, first, assign_dst, polarity, op_W, op_b, N_T, E_AS);
  // 6) TransportConv edge phase
  transport_edges<<<(E_TR + TPB - 1) / TPB, TPB, 0, stream>>>(
      x_torus, tr_ei, gate_W, gate_b, S, cnt, E_TR);
  // 7) TransportConv finalize (mean + residual)
  transport_finalize<<<(N_TOR + TPB - 1) / TPB, TPB, 0, stream>>>(
      x_torus, S, cnt, xtor_out, N_TOR);
}